// GraphCluster_18854906429737
// MI455X (gfx1250) — compile-verified
//
#include <hip/hip_runtime.h>
#include <hip/hip_bf16.h>

typedef float v2f __attribute__((ext_vector_type(2)));
typedef float v8f __attribute__((ext_vector_type(8)));

#define NNODES 100000

// ---------------------------------------------------------------- utilities
__global__ void zero_f32(float* __restrict__ p, long long n) {
  long long i = (long long)blockIdx.x * blockDim.x + threadIdx.x;
  long long stride = (long long)gridDim.x * blockDim.x;
  for (; i < n; i += stride) p[i] = 0.0f;
}

__global__ void degree_kernel(const int* __restrict__ dst, float* __restrict__ cnt, int nedges) {
  int e = blockIdx.x * blockDim.x + threadIdx.x;
  if (e < nedges) atomicAdd(&cnt[dst[e]], 1.0f);
}

__global__ void invcnt_kernel(const float* __restrict__ cnt, float* __restrict__ inv, int n) {
  int i = blockIdx.x * blockDim.x + threadIdx.x;
  if (i < n) { float c = cnt[i]; inv[i] = (c > 0.0f) ? (1.0f / c) : 0.0f; }
}

// ------------------------------------------------------- edge scatter (mean numerator)
// Consecutive lanes cover consecutive channel chunks of the same edge -> coalesced
// float4 gather + 4x global_atomic_add_f32 into the L2-resident msg buffer.
template<int DIN>
__global__ void scatter_kernel(const float* __restrict__ h, const int* __restrict__ src,
                               const int* __restrict__ dst, float* __restrict__ msg, int nedges) {
  constexpr int chunks = DIN >> 2;
  long long tid = (long long)blockIdx.x * blockDim.x + threadIdx.x;
  long long total = (long long)nedges * chunks;
  if (tid >= total) return;
  int e  = (int)(tid / chunks);
  int c4 = ((int)(tid % chunks)) << 2;
  int s = __ldg(src + e);
  int d = __ldg(dst + e);
  const float4 v = *reinterpret_cast<const float4*>(h + (size_t)s * DIN + c4);
  float* p = msg + (size_t)d * DIN + c4;
  atomicAdd(p + 0, v.x);
  atomicAdd(p + 1, v.y);
  atomicAdd(p + 2, v.z);
  atomicAdd(p + 3, v.w);
}

// ------------------------------------------------------- fused SAGEConv GEMM (f32 WMMA)
// One wave computes the full 16 x DOUT row-block:
//   out = (msg*inv) @ Wl + h @ Wr + b   [+ relu]
// A fragment (16x4 f32, per ISA): lanes 0-15 hold K=k,k+1 ; lanes 16-31 hold K=k+2,k+3.
// B fragment (4x16 f32): N striped across lanes, same K split across halves.
// C/D (16x16 f32, 8 VGPRs): VGPR v -> M = v + 8*(lane/16), N = lane%16 + 16*t.
// DIN/DOUT are compile-time so every load/store offset is an immediate.
template<int DIN, int DOUT, int RELU>
__global__ void __launch_bounds__(256) sage_wmma_kernel(
    const float* __restrict__ h, const float* __restrict__ msg, const float* __restrict__ invc,
    const float* __restrict__ Wl, const float* __restrict__ Wr, const float* __restrict__ bias,
    float* __restrict__ out)
{
  constexpr int NT = (DOUT + 15) / 16;     // N tiles handled by this wave
  int wave = (int)((blockIdx.x * (unsigned)blockDim.x + threadIdx.x) >> 5);
  int lane = (int)(threadIdx.x & 31u);
  if (wave >= NNODES / 16) return;         // whole-wave exit: EXEC stays all-1 for WMMA

  const int tm   = wave * 16;
  const int half = lane >> 4;              // 0: K=k,k+1   1: K=k+2,k+3
  const int lr   = lane & 15;
  const int row  = tm + lr;                // A-fragment row (same for both halves)

  const float inv = invc[row];
  // A-fragment base pointers, pre-biased by the half's K offset:
  const float* __restrict__ hrow = h   + (size_t)row * DIN + 2 * half;
  const float* __restrict__ mrow = msg + (size_t)row * DIN + 2 * half;

  // per-(tile,lane) W base pointers: column clamped for the partial tile
  const float* wlb[NT];
  const float* wrb[NT];
  bool ok[NT];
#pragma unroll
  for (int t = 0; t < NT; ++t) {
    const int col = t * 16 + lr;
    ok[t] = (col < DOUT);
    const int cc = ok[t] ? col : (DOUT - 1);
    wlb[t] = Wl + (size_t)(2 * half) * DOUT + cc;
    wrb[t] = Wr + (size_t)(2 * half) * DOUT + cc;
  }

  v8f c[NT];
#pragma unroll
  for (int t = 0; t < NT; ++t) {
    v8f z = {0.f, 0.f, 0.f, 0.f, 0.f, 0.f, 0.f, 0.f};
    c[t] = z;
  }

#pragma unroll
  for (int k = 0; k < DIN; k += 4) {
    v2f am, ah;
    am.x = mrow[k] * inv;  am.y = mrow[k + 1] * inv;   // immediate offsets
    ah.x = hrow[k];        ah.y = hrow[k + 1];
#pragma unroll
    for (int t = 0; t < NT; ++t) {
      v2f bl, br;
      float bl0 = wlb[t][(size_t)k * DOUT];
      float bl1 = wlb[t][(size_t)(k + 1) * DOUT];
      float br0 = wrb[t][(size_t)k * DOUT];
      float br1 = wrb[t][(size_t)(k + 1) * DOUT];
      if ((t + 1) * 16 <= DOUT) {          // full tile: no masking needed
        bl.x = bl0; bl.y = bl1; br.x = br0; br.y = br1;
      } else {                             // partial tile (DOUT=18)
        bl.x = ok[t] ? bl0 : 0.f; bl.y = ok[t] ? bl1 : 0.f;
        br.x = ok[t] ? br0 : 0.f; br.y = ok[t] ? br1 : 0.f;
      }
      c[t] = __builtin_amdgcn_wmma_f32_16x16x4_f32(false, am, false, bl, (short)0, c[t], false, false);
      c[t] = __builtin_amdgcn_wmma_f32_16x16x4_f32(false, ah, false, br, (short)0, c[t], false, false);
    }
  }

  // epilogue: bias + relu + store (immediate offsets off one base pointer)
  float* __restrict__ outp = out + (size_t)(tm + 8 * half) * DOUT + lr;
#pragma unroll
  for (int t = 0; t < NT; ++t) {
    const float bv = bias[ok[t] ? (t * 16 + lr) : (DOUT - 1)];
    if ((t + 1) * 16 <= DOUT) {
#pragma unroll
      for (int v = 0; v < 8; ++v) {
        float val = c[t][v] + bv;
        if (RELU) val = fmaxf(val, 0.0f);
        outp[(size_t)v * DOUT + t * 16] = val;
      }
    } else if (ok[t]) {
#pragma unroll
      for (int v = 0; v < 8; ++v) {
        float val = c[t][v] + bv;
        if (RELU) val = fmaxf(val, 0.0f);
        outp[(size_t)v * DOUT + t * 16] = val;
      }
    }
  }
}

// ---------------------------------------------------------------- driver
extern "C" void kernel_launch(void* const* d_in, const int* in_sizes, int n_in,
                              void* d_out, int out_size, void* d_ws, size_t ws_size,
                              hipStream_t stream) {
  const float* x  = (const float*)d_in[0];
  const int*   ei = (const int*)d_in[1];
  const int nedges = in_sizes[1] / 2;
  const int* src = ei;
  const int* dst = ei + nedges;

  static const int dins[8]  = {128, 64, 64, 64, 64, 64, 64, 64};
  static const int douts[8] = { 64, 64, 64, 64, 64, 64, 64, 18};

  const float *Wl[8], *Wr[8], *B[8];
  if (n_in >= 26) {                       // tuples expanded into separate inputs
    for (int i = 0; i < 8; ++i) {
      Wl[i] = (const float*)d_in[2 + i];
      Wr[i] = (const float*)d_in[10 + i];
      B[i]  = (const float*)d_in[18 + i];
    }
  } else {                                // tuples flattened into single buffers
    const float* wl0 = (const float*)d_in[2];
    const float* wr0 = (const float*)d_in[3];
    const float* b0  = (const float*)d_in[4];
    size_t wo = 0, bo = 0;
    for (int i = 0; i < 8; ++i) {
      Wl[i] = wl0 + wo; Wr[i] = wr0 + wo; B[i] = b0 + bo;
      wo += (size_t)dins[i] * douts[i]; bo += (size_t)douts[i];
    }
  }

  // workspace carve-up (~103 MB)
  float* ws  = (float*)d_ws;
  float* msg = ws;                                  // NNODES*128
  float* hA  = msg + (size_t)NNODES * 128;          // NNODES*64
  float* hB  = hA  + (size_t)NNODES * 64;           // NNODES*64
  float* cnt = hB  + (size_t)NNODES * 64;           // NNODES
  float* inv = cnt + NNODES;                        // NNODES

  // degrees (shared by all layers)
  zero_f32<<<1024, 256, 0, stream>>>(cnt, (long long)NNODES);
  degree_kernel<<<(nedges + 255) / 256, 256, 0, stream>>>(dst, cnt, nedges);
  invcnt_kernel<<<(NNODES + 255) / 256, 256, 0, stream>>>(cnt, inv, NNODES);

  const int waves   = NNODES / 16;                  // 6250 row-block waves
  const int gblocks = (waves * 32 + 255) / 256;

  const float* hin = x;
  float* hout = hA;
  for (int L = 0; L < 8; ++L) {
    const int din = dins[L];
    const long long msgN = (long long)NNODES * din;
    zero_f32<<<4096, 256, 0, stream>>>(msg, msgN);

    const long long total = (long long)nedges * (din >> 2);
    const int sblocks = (int)((total + 255) / 256);
    float* dbuf = (L == 7) ? (float*)d_out : hout;

    if (L == 0) {
      scatter_kernel<128><<<sblocks, 256, 0, stream>>>(hin, src, dst, msg, nedges);
      sage_wmma_kernel<128, 64, 1><<<gblocks, 256, 0, stream>>>(hin, msg, inv, Wl[L], Wr[L], B[L], dbuf);
    } else if (L < 7) {
      scatter_kernel<64><<<sblocks, 256, 0, stream>>>(hin, src, dst, msg, nedges);
      sage_wmma_kernel<64, 64, 1><<<gblocks, 256, 0, stream>>>(hin, msg, inv, Wl[L], Wr[L], B[L], dbuf);
    } else {
      scatter_kernel<64><<<sblocks, 256, 0, stream>>>(hin, src, dst, msg, nedges);
      sage_wmma_kernel<64, 18, 0><<<gblocks, 256, 0, stream>>>(hin, msg, inv, Wl[L], Wr[L], B[L], dbuf);
    }
    hin  = dbuf;
    hout = (hout == hA) ? hB : hA;
  }
}